// BWModel_82669530513757
// MI455X (gfx1250) — compile-verified
//
#include <hip/hip_runtime.h>
#include <hip/hip_bf16.h>
#include <math.h>

typedef __attribute__((ext_vector_type(2))) float v2f;
typedef __attribute__((ext_vector_type(8))) float v8f;

#define NZ_HALF 13      // z slices 0..12
#define NR      35      // anchor radii
#define NRHO    101     // integration points
#define KP      104     // padded K (26 x 4)
#define NPAD    48      // padded N (3 tiles of 16)
#define VOL     15625   // 25*25*25
#define NCHAN   1024    // 16*64

// Abramowitz & Stegun / NR rational approximation of J0 (f32, matches reference)
__device__ __forceinline__ float j0f_dev(float x) {
    float ax = fabsf(x);
    if (ax <= 8.0f) {
        float y = x * x;
        float num = 57568490574.0f + y * (-13362590354.0f + y * (651619640.7f +
                    y * (-11214424.18f + y * (77392.33017f + y * (-184.9052456f)))));
        float den = 57568490411.0f + y * (1029532985.0f + y * (9494680.718f +
                    y * (59272.64853f + y * (267.8532712f + y))));
        return num / den;
    } else {
        float z  = 8.0f / ax;
        float y  = z * z;
        float xx = ax - 0.785398164f;
        float p1 = 1.0f + y * (-0.1098628627e-2f + y * (0.2734510407e-4f +
                   y * (-0.2073370639e-5f + y * 0.2093887211e-6f)));
        float p2 = -0.1562499995e-1f + y * (0.1430488765e-3f + y * (-0.6911147651e-5f +
                   y * (0.7621095161e-6f + y * (-0.934935152e-7f))));
        return sqrtf(0.636619772f / ax) * (cosf(xx) * p1 - z * sinf(xx) * p2);
    }
}

__global__ __launch_bounds__(256)
void bornwolf_psf_kernel(const float* __restrict__ params, float* __restrict__ out) {
    __shared__ __align__(16) float sAcos[16 * KP];    // [m][k] f32
    __shared__ __align__(16) float sAsin[16 * KP];    // [m][k] f32
    __shared__ __align__(16) float sB[NPAD * KP];     // [n][k] f32 (K-contiguous per column)
    __shared__ float sRe[16 * NPAD];                  // pr, later |.|^2 plane
    __shared__ float sIm[16 * NPAD];                  // pi
    __shared__ float sZsum[NR];
    __shared__ float sRed[256];

    const int cid  = blockIdx.x;          // channel index b*64+c
    const int tid  = threadIdx.x;
    const int lane = tid & 31;
    const int wid  = tid >> 5;

    const float lam = fabsf(params[cid * 2 + 0]);
    const float n   = fabsf(params[cid * 2 + 1]);
    const float k   = 6.283185307179586f / lam;
    const float n2  = n * n;

    // ---- Phase 1: build A (cos/sin of phase) and B (J0 * rho * trapezoid weight), f32 ----
    for (int idx = tid; idx < 16 * KP; idx += 256) {
        int m = idx / KP, i = idx - m * KP;
        float c = 0.0f, s = 0.0f;
        if (m < NZ_HALF && i < NRHO) {
            float rho = (float)i * 0.01f;
            float w = 0.5f * k * rho * rho * (float)m * n2;   // z = m * pixel_size_z(=1)
            c = cosf(w);
            s = sinf(w);
        }
        sAcos[idx] = c;
        sAsin[idx] = s;
    }
    for (int idx = tid; idx < NPAD * KP; idx += 256) {
        int nn = idx / KP, i = idx - nn * KP;
        float v = 0.0f;
        if (nn < NR && i < NRHO) {
            float rho = (float)i * 0.01f;
            float wt  = 0.01f * ((i == 0 || i == NRHO - 1) ? 0.5f : 1.0f);
            float r   = (float)nn * 0.5f;                      // _R[nn] = nn / OS
            v = j0f_dev(k * n * r * rho) * rho * wt;
        }
        sB[idx] = v;
    }
    __syncthreads();

    // ---- Phase 2: 6 wave-level GEMM tiles: (re|im) x (3 N-tiles), 26 x V_WMMA_F32_16X16X4_F32 ----
    if (wid < 6) {
        const int sel  = wid & 1;                 // 0: cos -> pr, 1: sin -> pi
        const int tile = wid >> 1;                // N tile 0..2
        const float* A = sel ? sAsin : sAcos;
        const int mrow = lane & 15;
        const int ncol = tile * 16 + (lane & 15);
        const int kl   = (lane < 16) ? 0 : 2;     // half-wave K offset (A and B identical here)
        v8f acc = {};
        #pragma unroll
        for (int ks = 0; ks < 26; ks++) {
            const int kb = ks * 4;
            // A 16x4 f32 fragment: VGPR0/1 = K = kb+kl, kb+kl+1 for row M=mrow
            v2f af = *(const v2f*)&A[mrow * KP + kb + kl];
            // B 4x16 f32 fragment: VGPR v -> K = kb+kl+v, N = ncol (K-contiguous in LDS)
            v2f bf = *(const v2f*)&sB[ncol * KP + kb + kl];
            acc = __builtin_amdgcn_wmma_f32_16x16x4_f32(
                false, af, false, bf, (short)0, acc, false, false);
        }
        // D layout: VGPR v -> M = v + (lane<16 ? 0 : 8), N = ncol
        float* dst = sel ? sIm : sRe;
        const int mbase = (lane < 16) ? 0 : 8;
        #pragma unroll
        for (int v = 0; v < 8; v++) dst[(v + mbase) * NPAD + ncol] = acc[v];
    }
    __syncthreads();

    // ---- Phase 3: plane = pr^2 + pi^2 (overwrite sRe), then z-collapsed radial sums ----
    for (int idx = tid; idx < NZ_HALF * NR; idx += 256) {
        int m = idx / NR, r = idx - m * NR;
        float re = sRe[m * NPAD + r], im = sIm[m * NPAD + r];
        sRe[m * NPAD + r] = re * re + im * im;
    }
    __syncthreads();
    if (tid < NR) {
        float z = sRe[tid];                        // m=0 appears once (z=12)
        for (int m = 1; m < NZ_HALF; m++) z += 2.0f * sRe[m * NPAD + tid];
        sZsum[tid] = z;
    }
    __syncthreads();

    // ---- Phase 4: normalization total = sum over (y,x) of interpolated zsum ----
    float local = 0.0f;
    for (int p = tid; p < 625; p += 256) {
        int y = p / 25, x = p - y * 25;
        float dy = (float)y - 12.0f, dx = (float)x - 12.0f;
        float rp = sqrtf(dx * dx + dy * dy);
        float fi = floorf(rp * 2.0f);
        int   i1 = (int)fi, i2 = i1 + 1;
        float d1 = rp * 2.0f - fi;                 // (rPixel - _R[i1]) * OS
        local += sZsum[i2] * d1 + sZsum[i1] * (1.0f - d1);
    }
    sRed[tid] = local;
    __syncthreads();
    for (int off = 128; off >= 1; off >>= 1) {
        if (tid < off) sRed[tid] += sRed[tid + off];
        __syncthreads();
    }
    const float inv = 1.0f / sRed[0];

    // ---- Phase 5: coalesced write of the 25^3 volume ----
    float* o = out + (size_t)cid * VOL;
    for (int i = tid; i < VOL; i += 256) {
        int z   = i / 625;
        int rem = i - z * 625;
        int y   = rem / 25;
        int x   = rem - y * 25;
        int m   = (z >= 12) ? (z - 12) : (12 - z);  // reflect pad
        float dy = (float)y - 12.0f, dx = (float)x - 12.0f;
        float rp = sqrtf(dx * dx + dy * dy);
        float fi = floorf(rp * 2.0f);
        int   i1 = (int)fi, i2 = i1 + 1;
        float d1 = rp * 2.0f - fi;
        float val = sRe[m * NPAD + i2] * d1 + sRe[m * NPAD + i1] * (1.0f - d1);
        o[i] = val * inv;
    }
}

extern "C" void kernel_launch(void* const* d_in, const int* in_sizes, int n_in,
                              void* d_out, int out_size, void* d_ws, size_t ws_size,
                              hipStream_t stream) {
    (void)in_sizes; (void)n_in; (void)out_size; (void)d_ws; (void)ws_size;
    const float* params = (const float*)d_in[0];   // (16, 64, 2) f32
    float* out = (float*)d_out;                    // (16, 64, 25, 25, 25) f32
    bornwolf_psf_kernel<<<dim3(NCHAN), dim3(256), 0, stream>>>(params, out);
}